// cd_func_9062380995248
// MI455X (gfx1250) — compile-verified
//
#include <hip/hip_runtime.h>

typedef __attribute__((ext_vector_type(2))) float v2f;
typedef __attribute__((ext_vector_type(8))) float v8f;

#define BATCH 16
#define NPTS  4096
#define NTILES (NPTS / 16)      // 256 tiles of 16 points
#define WAVES_PER_BLOCK 8       // 256 threads = 8 wave32

__global__ void chamfer_zero(float* out) { out[0] = 0.0f; }

// One wave per (batch, 16-row tile of `apts`). Loops over all 16-col tiles of
// `bpts`, computing the 16x16 squared-distance tile with a single
// V_WMMA_F32_16X16X4_F32:
//   A row (K=0..3) = (-2ax, -2ay, -2az, 1    )
//   B col (K=0..3) = ( bx,   by,   bz,  |b|^2)
//   C (loop-inv)   = |a|^2 (per row)
//   D = A*B + C    = |a|^2 + |b|^2 - 2 a.b = d^2
// Running elementwise min over tiles (clamp to 0 deferred to after the loop:
// min_i max(x_i,0) == max(min_i x_i, 0)), then butterfly-min across the 16
// lanes of each half-wave (C/D layout: VGPR r = row r / r+8, col = lane&15),
// mean over the 16 rows, atomicAdd into the scalar loss.
__global__ __launch_bounds__(256) void chamfer_dir(const float* __restrict__ apts,
                                                   const float* __restrict__ bpts,
                                                   float* __restrict__ out) {
  const int lane = threadIdx.x & 31;
  const int wave = threadIdx.x >> 5;
  const int gw   = blockIdx.x * WAVES_PER_BLOCK + wave;  // 0 .. BATCH*NTILES-1
  const int b    = gw >> 8;        // gw / NTILES
  const int tile = gw & 255;       // gw % NTILES
  const int half = lane >> 4;      // 0: K=0,1 half   1: K=2,3 half
  const int l16  = lane & 15;

  const float* __restrict__ abase = apts + (size_t)b * NPTS * 3;
  const float* __restrict__ bbase = bpts + (size_t)b * NPTS * 3;

  // ---- A matrix (constant across the whole loop) ----
  const int arow = tile * 16 + l16;
  const float ax = abase[arow * 3 + 0];
  const float ay = abase[arow * 3 + 1];
  const float az = abase[arow * 3 + 2];
  const float a2 = ax * ax + ay * ay + az * az;  // lane l holds |a|^2 of row l&15
  v2f A;
  A.x = (half == 0) ? (-2.0f * ax) : (-2.0f * az);  // K=0 / K=2
  A.y = (half == 0) ? (-2.0f * ay) : 1.0f;          // K=1 / K=3

  // ---- C matrix: |a|^2 per row, loop-invariant ----
  // C/D element (VGPR r, lane l) is row r + half*8 -> fetch that row's |a|^2
  // from the lane that computed it (lane r + half*8).
  v8f cA;
#pragma unroll
  for (int r = 0; r < 8; ++r) cA[r] = __shfl(a2, r + half * 8, 32);

  // running per-element min in C/D layout (unclamped)
  v8f m;
#pragma unroll
  for (int r = 0; r < 8; ++r) m[r] = 3.4e38f;

#pragma unroll 4
  for (int t = 0; t < NTILES; ++t) {
    const int col = t * 16 + l16;
    const float t0 = bbase[col * 3 + 0];
    const float t1 = bbase[col * 3 + 1];
    const float t2 = bbase[col * 3 + 2];
    const float b2 = t0 * t0 + t1 * t1 + t2 * t2;

    v2f Bm;
    Bm.x = (half == 0) ? t0 : t2;     // K=0 / K=2
    Bm.y = (half == 0) ? t1 : b2;     // K=1 / K=3

    // d[r] = |a|^2 + |b|^2 - 2 a.b  for (row = tile*16 + r + half*8, col)
    v8f d = __builtin_amdgcn_wmma_f32_16x16x4_f32(
        /*neg_a=*/false, A, /*neg_b=*/false, Bm,
        /*c_mod=*/(short)0, cA, /*reuse_a=*/false, /*reuse_b=*/false);

#pragma unroll
    for (int r = 0; r < 8; ++r) m[r] = fminf(m[r], d[r]);
  }

  // deferred clamp (== per-tile max(d2,0) by monotonicity), then min over the
  // 16 columns via butterfly within each 16-lane half
#pragma unroll
  for (int r = 0; r < 8; ++r) {
    float v = fmaxf(m[r], 0.0f);
#pragma unroll
    for (int s = 1; s < 16; s <<= 1) {
      v = fminf(v, __shfl_xor(v, s, 32));
    }
    m[r] = v;  // all lanes of each half now hold min for row r (+ half*8)
  }

  // sum the 8 row-mins this half owns, combine halves, scale by 1/N, accumulate
  float part = 0.0f;
#pragma unroll
  for (int r = 0; r < 8; ++r) part += m[r];
  float other = __shfl_xor(part, 16, 32);
  if (lane == 0) {
    atomicAdd(out, (part + other) * (1.0f / (float)NPTS));
  }
}

extern "C" void kernel_launch(void* const* d_in, const int* in_sizes, int n_in,
                              void* d_out, int out_size, void* d_ws, size_t ws_size,
                              hipStream_t stream) {
  const float* src = (const float*)d_in[0];  // [16, 4096, 3]
  const float* tgt = (const float*)d_in[1];  // [16, 4096, 3]
  float* out = (float*)d_out;                // scalar loss

  chamfer_zero<<<1, 1, 0, stream>>>(out);

  const int total_waves = BATCH * NTILES;                 // 4096 per direction
  const int blocks = total_waves / WAVES_PER_BLOCK;       // 512
  chamfer_dir<<<blocks, 256, 0, stream>>>(src, tgt, out); // sum(mean(cham_x))
  chamfer_dir<<<blocks, 256, 0, stream>>>(tgt, src, out); // sum(mean(cham_y))
}